// PolygonFeatureGathering_26938034880614
// MI455X (gfx1250) — compile-verified
//
#include <hip/hip_runtime.h>

#define N_NODES 50000
#define N_EDGES 800000
#define IN_DIM  128
#define HID     128
#define OUT_DIM 64

typedef __attribute__((ext_vector_type(2))) float v2f;
typedef __attribute__((ext_vector_type(8))) float v8f;

// ---------------------------------------------------------------- utilities
__global__ void zero_kernel(float* __restrict__ p, int n) {
  int i = blockIdx.x * blockDim.x + threadIdx.x;
  int stride = gridDim.x * blockDim.x;
  for (; i < n; i += stride) p[i] = 0.0f;
}

// deg[col[e]] += 1 for every edge (self-loop handled in dinv_kernel)
__global__ void degree_kernel(const long long* __restrict__ col,
                              float* __restrict__ deg, int e) {
  int i = blockIdx.x * blockDim.x + threadIdx.x;
  if (i < e) atomicAdd(&deg[(int)col[i]], 1.0f);
}

// dinv = rsqrt(deg + 1)   (+1 = self-loop; always > 0)
__global__ void dinv_kernel(const float* __restrict__ deg,
                            float* __restrict__ dinv, int n) {
  int i = blockIdx.x * blockDim.x + threadIdx.x;
  if (i < n) dinv[i] = rsqrtf(deg[i] + 1.0f);
}

// ------------------------------------------------- fp32 WMMA GEMM (K = 128)
// C[rows, NCOLS] = A[rows,128] @ B[128,NCOLS] (+ bias)
// One wave computes a 16 x NCOLS output stripe via V_WMMA_F32_16X16X4_F32.
template <int NCOLS>
__global__ void wmma_gemm_f32(const float* __restrict__ A,
                              const float* __restrict__ B,
                              float* __restrict__ C,
                              const float* __restrict__ bias,
                              int rowTiles) {
  constexpr int NT = NCOLS / 16;
  const int lane = threadIdx.x & 31;
  const int wave = threadIdx.x >> 5;
  const int tile = blockIdx.x * (blockDim.x >> 5) + wave;
  if (tile >= rowTiles) return;            // uniform per wave: EXEC stays all-1s

  const int m0    = tile * 16;
  const int m     = lane & 15;             // row within tile (A), col within tile (B/C)
  const int khalf = (lane >> 4) << 1;      // lanes 0-15 -> K+0/K+1, lanes 16-31 -> K+2/K+3

  v8f acc[NT];
#pragma unroll
  for (int t = 0; t < NT; ++t) acc[t] = (v8f){0.f, 0.f, 0.f, 0.f, 0.f, 0.f, 0.f, 0.f};

  for (int k = 0; k < 128; k += 4) {
    // A fragment: A[m0+m][k+khalf .. k+khalf+1]
    const float* ap = A + (size_t)(m0 + m) * 128 + (k + khalf);
    v2f a;
    a.x = ap[0];
    a.y = ap[1];
#pragma unroll
    for (int t = 0; t < NT; ++t) {
      // B fragment: B[k+khalf][n], B[k+khalf+1][n]   with n = t*16 + m
      const float* bp = B + (size_t)(k + khalf) * NCOLS + t * 16 + m;
      v2f b;
      b.x = bp[0];
      b.y = bp[NCOLS];
      acc[t] = __builtin_amdgcn_wmma_f32_16x16x4_f32(
          /*neg_a=*/false, a, /*neg_b=*/false, b,
          /*c_mod=*/(short)0, acc[t], /*reuse_a=*/false, /*reuse_b=*/false);
    }
  }

  // C/D layout: VGPR v -> row m0 + v + (lane>=16 ? 8 : 0), col = t*16 + (lane&15)
  const int srow = m0 + ((lane >> 4) << 3);
#pragma unroll
  for (int t = 0; t < NT; ++t) {
    const float bv = bias ? bias[t * 16 + m] : 0.0f;
#pragma unroll
    for (int v = 0; v < 8; ++v) {
      C[(size_t)(srow + v) * NCOLS + t * 16 + m] = acc[t][v] + bv;
    }
  }
}

// ------------------------------------------------------------- edge scatter
// One wave per edge: agg[col] += x[row] * dinv[row]*dinv[col]  (128 feats)
__global__ void scatter_kernel(const long long* __restrict__ erow,
                               const long long* __restrict__ ecol,
                               const float* __restrict__ x,
                               const float* __restrict__ dinv,
                               float* __restrict__ agg, int e) {
  const int lane = threadIdx.x & 31;
  const int edge = (blockIdx.x * blockDim.x + threadIdx.x) >> 5;
  if (edge >= e) return;
  const int r = (int)erow[edge];       // scalar-broadcast loads (wave-uniform)
  const int c = (int)ecol[edge];
  const float norm = dinv[r] * dinv[c];
  const float* xs = x + (size_t)r * HID;
  float* ad = agg + (size_t)c * HID;
#pragma unroll
  for (int j = 0; j < HID; j += 32) {
    atomicAdd(&ad[j + lane], xs[j + lane] * norm);
  }
}

// ------------------------------------ self-loop + bias + ReLU (in place on agg)
__global__ void finalize_kernel(float* __restrict__ agg,
                                const float* __restrict__ x,
                                const float* __restrict__ dinv,
                                const float* __restrict__ b_gcn, int total) {
  int i = blockIdx.x * blockDim.x + threadIdx.x;
  if (i < total) {
    const int node = i >> 7;           // /128
    const int j = i & 127;
    const float d = dinv[node];
    const float v = agg[i] + x[i] * d * d + b_gcn[j];
    agg[i] = v > 0.0f ? v : 0.0f;
  }
}

// --------------------------------------------------------------------------
extern "C" void kernel_launch(void* const* d_in, const int* in_sizes, int n_in,
                              void* d_out, int out_size, void* d_ws, size_t ws_size,
                              hipStream_t stream) {
  const float*     h     = (const float*)d_in[0];      // [N, 128]
  const long long* edge  = (const long long*)d_in[1];  // [2, E] int64
  const float*     W_gcn = (const float*)d_in[2];      // [128, 128]
  const float*     b_gcn = (const float*)d_in[3];      // [128]
  const float*     W_fc  = (const float*)d_in[4];      // [128, 64]
  const float*     b_fc  = (const float*)d_in[5];      // [64]
  float*           out   = (float*)d_out;              // [N, 64]

  // Workspace layout (floats): x | agg | deg | dinv  (~51.6 MB)
  float* ws   = (float*)d_ws;
  float* x    = ws;                                   // N*128
  float* agg  = x + (size_t)N_NODES * HID;            // N*128
  float* deg  = agg + (size_t)N_NODES * HID;          // N
  float* dinv = deg + N_NODES;                        // N

  const long long* erow = edge;            // source nodes
  const long long* ecol = edge + N_EDGES;  // destination nodes

  // 1) zero accumulators
  zero_kernel<<<2048, 256, 0, stream>>>(agg, N_NODES * HID);
  zero_kernel<<<256, 256, 0, stream>>>(deg, N_NODES);

  // 2) in-degree (self-loop folded into dinv)
  degree_kernel<<<(N_EDGES + 255) / 256, 256, 0, stream>>>(ecol, deg, N_EDGES);
  dinv_kernel<<<(N_NODES + 255) / 256, 256, 0, stream>>>(deg, dinv, N_NODES);

  // 3) x = h @ W_gcn  (WMMA fp32; 3125 row tiles exactly cover N)
  const int rowTiles = N_NODES / 16;  // 3125
  wmma_gemm_f32<HID><<<(rowTiles + 7) / 8, 256, 0, stream>>>(
      h, W_gcn, x, nullptr, rowTiles);

  // 4) edge scatter (one wave per edge, float atomics into L2-resident agg)
  {
    const long long totalThreads = (long long)N_EDGES * 32;
    const int blocks = (int)((totalThreads + 255) / 256);
    scatter_kernel<<<blocks, 256, 0, stream>>>(erow, ecol, x, dinv, agg, N_EDGES);
  }

  // 5) agg = relu(agg + x*dinv^2 + b_gcn)
  finalize_kernel<<<(N_NODES * HID + 255) / 256, 256, 0, stream>>>(
      agg, x, dinv, b_gcn, N_NODES * HID);

  // 6) out = agg @ W_fc + b_fc  (WMMA fp32)
  wmma_gemm_f32<OUT_DIM><<<(rowTiles + 7) / 8, 256, 0, stream>>>(
      agg, W_fc, out, b_fc, rowTiles);
}